// HardDetectionModule_11708080849517
// MI455X (gfx1250) — compile-verified
//
#include <hip/hip_runtime.h>
#include <math.h>
#include <stdint.h>

#define BDIM 256
constexpr int B = 4, C = 256, H = 200, W = 320;
constexpr int TH = 32, TW = 64;
constexpr int HT = TH + 8;                    // 40 halo rows
constexpr int WT = TW + 8;                    // 72 halo cols
constexpr int TILES_W = (W + TW - 1) / TW;    // 5
constexpr int TILES_H = (H + TH - 1) / TH;    // 7
constexpr int HMS = 65;                       // padded hmax row stride (bank-conflict-free)
constexpr int NCHUNK_MAX = 4;
constexpr int BHW = B * H * W;                // 256000
constexpr int NSEG = WT / 4;                  // 18 x 16B segments per halo row
constexpr int NTASK = HT * NSEG;              // 720 copy tasks per tile

#if defined(__has_builtin)
# if __has_builtin(__builtin_amdgcn_global_load_async_to_lds_b128) && \
     __has_builtin(__builtin_amdgcn_s_wait_asynccnt)
#  define USE_ASYNC 1
# endif
#endif
#ifndef USE_ASYNC
# define USE_ASYNC 0
#endif

typedef int v4i __attribute__((ext_vector_type(4)));
typedef __attribute__((address_space(1))) v4i* g_v4i_ptr;
typedef __attribute__((address_space(3))) v4i* l_v4i_ptr;

__device__ __forceinline__ void copy16_to_lds(const float* g, float* l) {
#if USE_ASYNC
  // 16B global -> LDS async copy, tracked by ASYNCcnt.
  // Global: integer round-trip is safe (global == generic representation) and
  // drops constness. LDS: direct cast so clang emits a real addrspacecast.
  __builtin_amdgcn_global_load_async_to_lds_b128(
      (g_v4i_ptr)(uintptr_t)g, (l_v4i_ptr)l, 0, 0);
#else
  float4 v = *(const float4*)g;
  *(float4*)l = v;
#endif
}

__device__ __forceinline__ void wait_async_le3() {
#if USE_ASYNC
  __builtin_amdgcn_s_wait_asynccnt(3);   // keep the 3 just-issued loads in flight
#endif
}
__device__ __forceinline__ void wait_async_0() {
#if USE_ASYNC
  __builtin_amdgcn_s_wait_asynccnt(0);
#endif
}

__global__ __launch_bounds__(BDIM)
void hdm_main(const float* __restrict__ x,
              float* __restrict__ Mbuf, float* __restrict__ Sbuf,
              float* __restrict__ out, int nchunk) {
  __shared__ float xt[2][HT * WT];   // double-buffered halo tiles
  __shared__ float hm[HT * HMS];     // horizontal 9-window max
  __shared__ float dummy[4];         // sink for masked async lanes

  const int tid = threadIdx.x;
  const int tw = blockIdx.x % TILES_W;
  const int th = blockIdx.x / TILES_W;
  const int b  = blockIdx.y;
  const int chunk = blockIdx.z;
  const int cpc = C / nchunk;
  const int c0 = chunk * cpc;
  const int h0 = th * TH, w0 = tw * TW;
  const int gh0 = h0 - 4, gw0 = w0 - 4;

  const float NINF = -__builtin_inff();

  // Pre-fill both buffers with -inf: OOB halo regions are never overwritten,
  // giving the -inf padding semantics of the reference max_pool for free.
  for (int i = tid; i < 2 * HT * WT; i += BDIM) (&xt[0][0])[i] = NINF;
  __syncthreads();

  // Each thread owns one column x up-to-9 output rows for the channel scan.
  const int vcol = tid & 63;
  const int vblk = tid >> 6;           // 0..3, row block of 9

  float M[9], S[9];
#pragma unroll
  for (int t = 0; t < 9; ++t) { M[t] = NINF; S[t] = 0.0f; }

  auto load_tile = [&](int c, int bufsel) {
    const float* gplane = x + ((size_t)(b * C + c) * H) * W;
#pragma unroll
    for (int t = 0; t < 3; ++t) {
      int task = t * BDIM + tid;               // 0..767 (exactly 3 ops/thread)
      int r = task / NSEG;
      int seg = task - r * NSEG;
      int gh = gh0 + r;
      int gc = gw0 + seg * 4;                  // 16B-aligned (w0 % 64 == 0)
      bool ok = (task < NTASK) && ((unsigned)gh < (unsigned)H) &&
                (gc >= 0) && (gc <= W - 4);
      const float* g = ok ? (gplane + (size_t)gh * W + gc) : gplane;
      float* l = ok ? &xt[bufsel][r * WT + seg * 4] : dummy;
      copy16_to_lds(g, l);
    }
  };

  load_tile(c0, 0);
  for (int k = 0; k < cpc; ++k) {
    const int cur = k & 1;
    if (k + 1 < cpc) { load_tile(c0 + k + 1, cur ^ 1); wait_async_le3(); }
    else             { wait_async_0(); }
    __syncthreads();   // tile for channel c0+k resident in all waves

    // ---- horizontal 9-max (van Herk): 40 rows x 8 col-blocks of 9 ----
    for (int task = tid; task < HT * 8; task += BDIM) {
      const int cb = task & 7;
      const int r  = task >> 3;
      const float* row = &xt[cur][r * WT + cb * 9];
      float v[9], suf[9];
#pragma unroll
      for (int kk = 0; kk < 9; ++kk) v[kk] = row[kk];
      suf[8] = v[8];
#pragma unroll
      for (int kk = 7; kk >= 0; --kk) suf[kk] = fmaxf(v[kk], suf[kk + 1]);
      float* hrow = &hm[r * HMS];
      const int jb = cb * 9;                   // window start col (<= 63)
      hrow[jb] = suf[0];
      if (cb < 7) {
        float run = row[9];                    // next-block running prefix
#pragma unroll
        for (int t = 1; t < 9; ++t) {
          hrow[jb + t] = fmaxf(suf[t], run);
          if (t < 8) run = fmaxf(run, row[9 + t]);
        }
      }
    }
    __syncthreads();

    // ---- vertical 9-max (van Herk) + fused online channel-argmax ----
    {
      const int rb = vblk * 9;
      float v9[9], suf[9];
#pragma unroll
      for (int kk = 0; kk < 9; ++kk) v9[kk] = hm[(rb + kk) * HMS + vcol];
      suf[8] = v9[8];
#pragma unroll
      for (int kk = 7; kk >= 0; --kk) suf[kk] = fmaxf(v9[kk], suf[kk + 1]);
      const int nn = (vblk < 3) ? 8 : 4;       // next-block rows available
      float run = NINF;
#pragma unroll
      for (int t = 0; t < 9; ++t) {
        const int i = rb + t;                  // output row within tile
        if (i < TH) {
          float lmax = (t == 0) ? suf[0] : fmaxf(suf[t], run);
          float xv = xt[cur][(i + 4) * WT + vcol + 4];
          float contrib = (xv == lmax) ? xv : 0.0f;
          if (xv > M[t])      { M[t] = xv; S[t] = contrib; }
          else if (xv == M[t]) { S[t] += contrib; }
        }
        if (t < nn) run = fmaxf(run, hm[(rb + 9 + t) * HMS + vcol]);
      }
    }
    __syncthreads();   // tile 'cur' fully consumed before it is reloaded
  }

  // ---- epilogue: write per-chunk partials (or final when nchunk==1) ----
  {
    const int rb = vblk * 9;
#pragma unroll
    for (int t = 0; t < 9; ++t) {
      const int i = rb + t;
      const int gh = h0 + i;
      if (i < TH && gh < H) {
        const size_t idx = ((size_t)b * H + gh) * W + (w0 + vcol);
        if (nchunk > 1) {
          Mbuf[(size_t)chunk * BHW + idx] = M[t];
          Sbuf[(size_t)chunk * BHW + idx] = S[t];
        } else {
          out[idx] = S[t];
        }
      }
    }
  }
}

__global__ __launch_bounds__(BDIM)
void hdm_merge(const float* __restrict__ Mbuf, const float* __restrict__ Sbuf,
               float* __restrict__ out, int nchunk) {
  int idx = blockIdx.x * BDIM + threadIdx.x;
  if (idx >= BHW) return;
  float M = Mbuf[idx], S = Sbuf[idx];
  for (int k = 1; k < nchunk; ++k) {
    float m2 = Mbuf[(size_t)k * BHW + idx];
    float s2 = Sbuf[(size_t)k * BHW + idx];
    if (m2 > M)       { M = m2; S = s2; }
    else if (m2 == M) { S += s2; }
  }
  out[idx] = S;
}

extern "C" void kernel_launch(void* const* d_in, const int* in_sizes, int n_in,
                              void* d_out, int out_size, void* d_ws, size_t ws_size,
                              hipStream_t stream) {
  const float* x = (const float*)d_in[0];
  float* out = (float*)d_out;

  const size_t need = (size_t)2 * NCHUNK_MAX * BHW * sizeof(float);
  const int nchunk = (ws_size >= need) ? NCHUNK_MAX : 1;

  float* Mbuf = (float*)d_ws;
  float* Sbuf = Mbuf + (size_t)nchunk * BHW;

  dim3 grid(TILES_W * TILES_H, B, nchunk);
  hipLaunchKernelGGL(hdm_main, grid, dim3(BDIM), 0, stream, x, Mbuf, Sbuf, out, nchunk);
  if (nchunk > 1) {
    hipLaunchKernelGGL(hdm_merge, dim3((BHW + BDIM - 1) / BDIM), dim3(BDIM), 0, stream,
                       Mbuf, Sbuf, out, nchunk);
  }
}